// SAGEEncoder_86088324481902
// MI455X (gfx1250) — compile-verified
//
#include <hip/hip_runtime.h>
#include <math.h>

typedef float v2f __attribute__((ext_vector_type(2)));
typedef float v8f __attribute__((ext_vector_type(8)));

// ---------------------------------------------------------------------------
// fp32 WMMA: D(16x16) = A(16x4) * B(4x16) + C   (full precision matrix core)
// ---------------------------------------------------------------------------
__device__ __forceinline__ v8f wmma4(v2f a, v2f b, v8f c) {
  return __builtin_amdgcn_wmma_f32_16x16x4_f32(false, a, false, b, (short)0, c,
                                               false, false);
}

// 16-lane sum within each half-wave (xor masks 1,2,4,8 stay inside a half)
__device__ __forceinline__ float lane16_sum(float v) {
  v += __shfl_xor(v, 1, 32);
  v += __shfl_xor(v, 2, 32);
  v += __shfl_xor(v, 4, 32);
  v += __shfl_xor(v, 8, 32);
  return v;
}

__device__ __forceinline__ float silu(float v) {
  return v / (1.0f + __expf(-v));
}

// ---------------------------------------------------------------------------
// tiny utility kernels
// ---------------------------------------------------------------------------
__global__ void fill_kernel(float* __restrict__ p, float v, size_t n) {
  size_t i = (size_t)blockIdx.x * blockDim.x + threadIdx.x;
  if (i < n) p[i] = v;
}

__global__ void deg_kernel(const int* __restrict__ dst, float* __restrict__ deg,
                           int E) {
  int e = blockIdx.x * blockDim.x + threadIdx.x;
  if (e < E) atomicAdd(&deg[dst[e]], 1.0f);
}

// ---------------------------------------------------------------------------
// scatter: 16 lanes per edge, float4 per lane. max via sign-split int atomics
// (bit-exact for IEEE floats, init = -inf), mean-sum via global_atomic_add_f32
// ---------------------------------------------------------------------------
__global__ __launch_bounds__(256) void scatter_kernel(
    const float* __restrict__ x, const int* __restrict__ src,
    const int* __restrict__ dst, float* __restrict__ amax,
    float* __restrict__ asum, int E) {
  long long g = (long long)blockIdx.x * blockDim.x + threadIdx.x;
  int e = (int)(g >> 4);
  if (e >= E) return;
  int q = ((int)g & 15) * 4;
  int s = src[e], d = dst[e];
  const float4 v = *(const float4*)(x + (size_t)s * 64 + q);
  float* pm = amax + (size_t)d * 64 + q;
  float* ps = asum + (size_t)d * 64 + q;
  float vv[4] = {v.x, v.y, v.z, v.w};
#pragma unroll
  for (int i = 0; i < 4; ++i) {
    float val = vv[i];
    if (val >= 0.0f)
      atomicMax((int*)(pm + i), __float_as_int(val));
    else
      atomicMin((unsigned int*)(pm + i), (unsigned int)__float_as_int(val));
    atomicAdd(ps + i, val);
  }
}

// ---------------------------------------------------------------------------
// adapter: out[N,64] = x @ W^T + b   (WMMA, K=64)
// block = 256 (8 waves), 128 nodes/block, B staged in LDS [64][66]
// ---------------------------------------------------------------------------
__global__ __launch_bounds__(256) void adapter_kernel(
    const float* __restrict__ x, const float* __restrict__ w,
    const float* __restrict__ bias, float* __restrict__ out, int N) {
  __shared__ float Bs[64 * 66];
  int tid = threadIdx.x;
  for (int idx = tid; idx < 64 * 64; idx += 256) {
    int j = idx >> 6, k = idx & 63;
    Bs[k * 66 + j] = w[idx];  // B[k][j] = W[j][k]
  }
  __syncthreads();

  int wave = tid >> 5, lane = tid & 31;
  int half = lane >> 4, n = lane & 15;
  int m0 = blockIdx.x * 128 + wave * 16;
  int rA = m0 + n;
  if (rA > N - 1) rA = N - 1;
  const float* xp = x + (size_t)rA * 64;

  v8f acc[4] = {};
#pragma unroll
  for (int kk = 0; kk < 16; ++kk) {
    int c = kk * 4 + half * 2;
    v2f a;
    a.x = xp[c];
    a.y = xp[c + 1];
#pragma unroll
    for (int t = 0; t < 4; ++t) {
      v2f b;
      b.x = Bs[(kk * 4 + half * 2) * 66 + t * 16 + n];
      b.y = Bs[(kk * 4 + half * 2 + 1) * 66 + t * 16 + n];
      acc[t] = wmma4(a, b, acc[t]);
    }
  }
  int rowbase = m0 + half * 8;
#pragma unroll
  for (int t = 0; t < 4; ++t) {
    float bv = bias[t * 16 + n];
#pragma unroll
    for (int r = 0; r < 8; ++r) {
      int row = rowbase + r;
      if (row < N) out[(size_t)row * 64 + t * 16 + n] = acc[t][r] + bv;
    }
  }
}

// ---------------------------------------------------------------------------
// conv update: out = silu(LN(concat(max,mean) @ wl^T + bl + h @ wr^T) + res)
// A is 16x192 (amax' | amean | h), B (192x64) staged in LDS, 48 WMMA steps x 4
// N-subtiles per wave; LN fully in registers via half-wave shuffles.
// ---------------------------------------------------------------------------
__global__ __launch_bounds__(256) void conv_update_kernel(
    const float* __restrict__ amax, const float* __restrict__ asum,
    const float* __restrict__ deg, const float* __restrict__ xin,
    const float* __restrict__ wl,  // [64][128]
    const float* __restrict__ bl,  // [64]
    const float* __restrict__ wr,  // [64][64]
    const float* __restrict__ g, const float* __restrict__ b,
    const float* __restrict__ res, float* __restrict__ out, int N) {
  __shared__ float Bs[192 * 66];
  int tid = threadIdx.x;
  for (int idx = tid; idx < 64 * 128; idx += 256) {
    int j = idx >> 7, k = idx & 127;
    Bs[k * 66 + j] = wl[idx];  // B[k][j] = wl[j][k]
  }
  for (int idx = tid; idx < 64 * 64; idx += 256) {
    int j = idx >> 6, k = idx & 63;
    Bs[(128 + k) * 66 + j] = wr[idx];  // B[128+k][j] = wr[j][k]
  }
  __syncthreads();

  int wave = tid >> 5, lane = tid & 31;
  int half = lane >> 4, n = lane & 15;
  int m0 = blockIdx.x * 128 + wave * 16;
  int rA = m0 + n;
  if (rA > N - 1) rA = N - 1;

  float dg = deg[rA];
  float inv = 1.0f / fmaxf(dg, 1.0f);
  bool haved = dg > 0.0f;
  const float* amaxp = amax + (size_t)rA * 64;
  const float* asump = asum + (size_t)rA * 64;
  const float* xp = xin + (size_t)rA * 64;

  v8f acc[4] = {};
  auto step = [&](v2f a, int kg) {
#pragma unroll
    for (int t = 0; t < 4; ++t) {
      v2f bb;
      bb.x = Bs[(kg + half * 2) * 66 + t * 16 + n];
      bb.y = Bs[(kg + half * 2 + 1) * 66 + t * 16 + n];
      acc[t] = wmma4(a, bb, acc[t]);
    }
  };

#pragma unroll
  for (int kk = 0; kk < 16; ++kk) {  // max aggregation columns 0..63
    int c = kk * 4 + half * 2;
    v2f a;
    a.x = haved ? amaxp[c] : 0.0f;
    a.y = haved ? amaxp[c + 1] : 0.0f;
    step(a, kk * 4);
  }
#pragma unroll
  for (int kk = 0; kk < 16; ++kk) {  // mean aggregation columns 64..127
    int c = kk * 4 + half * 2;
    v2f a;
    a.x = asump[c] * inv;
    a.y = asump[c + 1] * inv;
    step(a, 64 + kk * 4);
  }
#pragma unroll
  for (int kk = 0; kk < 16; ++kk) {  // self features columns 128..191
    int c = kk * 4 + half * 2;
    v2f a;
    a.x = xp[c];
    a.y = xp[c + 1];
    step(a, 128 + kk * 4);
  }

  // epilogue: +bias, LayerNorm over 64, +residual, SiLU
  float blv[4], gv[4], bv[4];
#pragma unroll
  for (int t = 0; t < 4; ++t) {
    blv[t] = bl[t * 16 + n];
    gv[t] = g[t * 16 + n];
    bv[t] = b[t * 16 + n];
  }
  float s[8], sq[8];
#pragma unroll
  for (int r = 0; r < 8; ++r) { s[r] = 0.0f; sq[r] = 0.0f; }
#pragma unroll
  for (int t = 0; t < 4; ++t)
#pragma unroll
    for (int r = 0; r < 8; ++r) {
      float v = acc[t][r] + blv[t];
      acc[t][r] = v;
      s[r] += v;
      sq[r] += v * v;
    }
  int rowbase = m0 + half * 8;
#pragma unroll
  for (int r = 0; r < 8; ++r) {
    float ss = lane16_sum(s[r]);
    float qq = lane16_sum(sq[r]);
    float mean = ss * (1.0f / 64.0f);
    float var = qq * (1.0f / 64.0f) - mean * mean;
    float rstd = rsqrtf(var + 1e-5f);
    int row = rowbase + r;
    if (row < N) {
      const float* rp = res + (size_t)row * 64;
      float* op = out + (size_t)row * 64;
#pragma unroll
      for (int t = 0; t < 4; ++t) {
        float v = (acc[t][r] - mean) * rstd * gv[t] + bv[t] + rp[t * 16 + n];
        op[t * 16 + n] = silu(v);
      }
    }
  }
}

// ---------------------------------------------------------------------------
// MLP head: out = LN(silu(h @ W1^T + b1)) @ W2^T + b2  (fused, no N x 256 buf)
// block = 64 threads (2 waves), 32 nodes/block; h1 staged per-wave in LDS for
// the 256-wide LayerNorm; both GEMMs on fp32 WMMA.
// ---------------------------------------------------------------------------
__global__ __launch_bounds__(64) void mlp_kernel(
    const float* __restrict__ h, const float* __restrict__ w1,  // [256][64]
    const float* __restrict__ b1,                               // [256]
    const float* __restrict__ g, const float* __restrict__ bb,  // [256]
    const float* __restrict__ w2,                               // [64][256]
    const float* __restrict__ b2,                               // [64]
    float* __restrict__ out, int N) {
  __shared__ float Hs[2][16 * 258];
  __shared__ float Stat[2][32];  // [0..15]=mean, [16..31]=rstd per row
  int tid = threadIdx.x, wave = tid >> 5, lane = tid & 31;
  int half = lane >> 4, n = lane & 15;
  int m0 = blockIdx.x * 32 + wave * 16;
  int rA = m0 + n;
  if (rA > N - 1) rA = N - 1;
  const float* hp = h + (size_t)rA * 64;
  float* Hw = Hs[wave];

  // preload A fragments for K=64
  v2f af[16];
#pragma unroll
  for (int kk = 0; kk < 16; ++kk) {
    int c = kk * 4 + half * 2;
    af[kk].x = hp[c];
    af[kk].y = hp[c + 1];
  }

  float s[8], sq[8];
#pragma unroll
  for (int r = 0; r < 8; ++r) { s[r] = 0.0f; sq[r] = 0.0f; }

  // stage 1: 256 outputs in 4 groups of 4 subtiles
#pragma unroll
  for (int jg = 0; jg < 4; ++jg) {
    v8f acc[4] = {};
#pragma unroll
    for (int kk = 0; kk < 16; ++kk) {
      int k0 = kk * 4 + half * 2;
#pragma unroll
      for (int t = 0; t < 4; ++t) {
        int j = (jg * 4 + t) * 16 + n;
        v2f bf;
        bf.x = w1[(size_t)j * 64 + k0];
        bf.y = w1[(size_t)j * 64 + k0 + 1];
        acc[t] = wmma4(af[kk], bf, acc[t]);
      }
    }
#pragma unroll
    for (int t = 0; t < 4; ++t) {
      int j = (jg * 4 + t) * 16 + n;
      float biasj = b1[j];
#pragma unroll
      for (int r = 0; r < 8; ++r) {
        float v = silu(acc[t][r] + biasj);
        s[r] += v;
        sq[r] += v * v;
        Hw[(r + half * 8) * 258 + j] = v;
      }
    }
  }

  // row statistics for 256-wide LayerNorm
#pragma unroll
  for (int r = 0; r < 8; ++r) {
    float ss = lane16_sum(s[r]);
    float qq = lane16_sum(sq[r]);
    float mean = ss * (1.0f / 256.0f);
    float var = qq * (1.0f / 256.0f) - mean * mean;
    float rstd = rsqrtf(var + 1e-5f);
    if (n == 0) {
      Stat[wave][half * 8 + r] = mean;
      Stat[wave][16 + half * 8 + r] = rstd;
    }
  }
  __syncthreads();
  float meanL = Stat[wave][n];
  float rstdL = Stat[wave][16 + n];

  // stage 2: out = LN(h1) @ W2^T + b2, K=256
  v8f acc2[4] = {};
  for (int kk = 0; kk < 64; ++kk) {
    int k0 = kk * 4 + half * 2;
    v2f a;
    a.x = (Hw[n * 258 + k0] - meanL) * rstdL * g[k0] + bb[k0];
    a.y = (Hw[n * 258 + k0 + 1] - meanL) * rstdL * g[k0 + 1] + bb[k0 + 1];
#pragma unroll
    for (int t = 0; t < 4; ++t) {
      int j = t * 16 + n;
      v2f bf;
      bf.x = w2[(size_t)j * 256 + k0];
      bf.y = w2[(size_t)j * 256 + k0 + 1];
      acc2[t] = wmma4(a, bf, acc2[t]);
    }
  }
  int rowbase = m0 + half * 8;
#pragma unroll
  for (int t = 0; t < 4; ++t) {
    float b2v = b2[t * 16 + n];
#pragma unroll
    for (int r = 0; r < 8; ++r) {
      int row = rowbase + r;
      if (row < N) out[(size_t)row * 64 + t * 16 + n] = acc2[t][r] + b2v;
    }
  }
}

// ---------------------------------------------------------------------------
extern "C" void kernel_launch(void* const* d_in, const int* in_sizes, int n_in,
                              void* d_out, int out_size, void* d_ws,
                              size_t ws_size, hipStream_t stream) {
  const float* x = (const float*)d_in[0];
  const int* ei = (const int*)d_in[1];
  const float* wl = (const float*)d_in[2];
  const float* bl = (const float*)d_in[3];
  const float* wr = (const float*)d_in[4];
  const float* lng = (const float*)d_in[5];
  const float* lnb = (const float*)d_in[6];
  const float* aw = (const float*)d_in[7];
  const float* ab = (const float*)d_in[8];
  const float* w1 = (const float*)d_in[9];
  const float* b1 = (const float*)d_in[10];
  const float* mg = (const float*)d_in[11];
  const float* mb = (const float*)d_in[12];
  const float* w2 = (const float*)d_in[13];
  const float* b2 = (const float*)d_in[14];

  int N = in_sizes[0] / 64;
  int E = in_sizes[1] / 2;
  const int* src = ei;
  const int* dst = ei + E;

  float* ws = (float*)d_ws;
  size_t nh = (size_t)N * 64;
  float* xres = ws;
  float* amax = xres + nh;
  float* asum = amax + nh;
  float* hbuf = asum + nh;
  float* degb = hbuf + nh;
  float* outf = (float*)d_out;

  int nbNodes = (N + 127) / 128;
  int fillB = (int)((nh + 255) / 256);

  // degree (constant across layers)
  fill_kernel<<<(N + 255) / 256, 256, 0, stream>>>(degb, 0.0f, (size_t)N);
  deg_kernel<<<(E + 255) / 256, 256, 0, stream>>>(dst, degb, E);

  // residual adapter: xres = x @ adapter_w^T + adapter_b
  adapter_kernel<<<nbNodes, 256, 0, stream>>>(x, aw, ab, xres, N);

  const float* cur = x;
  for (int i = 0; i < 5; ++i) {
    float* nxt = (i % 2 == 0) ? hbuf : outf;
    fill_kernel<<<fillB, 256, 0, stream>>>(amax, -INFINITY, nh);
    fill_kernel<<<fillB, 256, 0, stream>>>(asum, 0.0f, nh);
    long long thr = (long long)E * 16;
    scatter_kernel<<<(int)((thr + 255) / 256), 256, 0, stream>>>(cur, src, dst,
                                                                 amax, asum, E);
    const float* resp = (i == 0) ? xres : cur;
    conv_update_kernel<<<nbNodes, 256, 0, stream>>>(
        amax, asum, degb, cur, wl + (size_t)i * 64 * 128, bl + i * 64,
        wr + (size_t)i * 64 * 64, lng + i * 64, lnb + i * 64, resp, nxt, N);
    cur = nxt;
  }
  // cur == hbuf after 5 layers; fused MLP head writes final output
  mlp_kernel<<<(N + 31) / 32, 64, 0, stream>>>(cur, w1, b1, mg, mb, w2, b2,
                                               outf, N);
}